// Renderer_8169027797498
// MI455X (gfx1250) — compile-verified
//
#include <hip/hip_runtime.h>
#include <math.h>

// ---------------------------------------------------------------------------
// MI455X (gfx1250) renderer.
// Rasterization = 512*512 pixels x 2000 faces of affine evaluations
// (w0, w1, z affine in pixel coords) -> V_WMMA_F32_16X16X4_F32:
//   A (16x4)  = per-pixel [X, Y, 1, 0]
//   B (4x16)  = per-face  [cx, cy, cc, 0] columns
//   D (16x16) = 16 pixels x 16 faces per WMMA (2048 FLOP/inst).
//  * Whole 96KB coefficient table staged once per block into LDS via
//    GLOBAL_LOAD_ASYNC_TO_LDS_B128 (+ s_wait_asynccnt), inner loop reads
//    become conflict-free ds_load_b64 (32 lanes x 8B = 64 banks exactly).
//  * Validity test collapsed to min3(u,w,z)>=0 && u+w<=1 && z<best.
// ---------------------------------------------------------------------------

#define IMG    512
#define NVERT  2048
#define NFACE  2000
#define NB     ((NFACE + 15) / 16)   // 125 batches of 16 faces
#define BIGF   1.0e10f
#define EPSA   1e-8f
#define BFLOATS (3 * NB * 64)        // 24000 floats = 96000 B of B-fragments

typedef float v2f __attribute__((ext_vector_type(2)));
typedef float v8f __attribute__((ext_vector_type(8)));

// GCC-style int4 to match the async-builtin's pointee type exactly
// (diagnostic: "__attribute__((__vector_size__(4 * sizeof(int)))) int __device__ *")
typedef int b128_t __attribute__((vector_size(16)));
typedef __attribute__((address_space(1))) b128_t* gv4i_p;  // global (AS1)
typedef __attribute__((address_space(3))) b128_t* lv4i_p;  // LDS (AS3)

#if __has_builtin(__builtin_amdgcn_global_load_async_to_lds_b128) && \
    __has_builtin(__builtin_amdgcn_s_wait_asynccnt)
#define USE_ASYNC_LDS 1
#else
#define USE_ASYNC_LDS 0
#endif

// ---------------- ws layout (floats, all offsets 16B-aligned) ---------------
// verts  (screen-space)      : [    0,  6144)
// nrm    (unit vtx normals)  : [ 6144, 12288)
// fn     (face normals)      : [12288, 18288)
// B fragments (contiguous)   : [18288, 18288 + 24000)   (Bw0 | Bw1 | Bz)

// --------------------------- projection ------------------------------------
__global__ void project_kernel(const float* __restrict__ vin,
                               const float* __restrict__ cam,
                               float* __restrict__ tv_out,
                               float* __restrict__ verts) {
  int v = blockIdx.x * blockDim.x + threadIdx.x;
  if (v >= NVERT) return;
  float x = vin[3 * v + 0], y = vin[3 * v + 1], z = vin[3 * v + 2];
  float s = cam[0], ox = cam[1], oy = cam[2];
  float tvx = s * (x + ox);
  float tvy = -(s * (y + oy));
  float tvz = 10.0f - s * z;
  tv_out[3 * v + 0] = tvx;
  tv_out[3 * v + 1] = tvy;
  tv_out[3 * v + 2] = tvz;
  verts[3 * v + 0] = -tvx;   // rasterizer verts = tv * [-1,-1,1]
  verts[3 * v + 1] = -tvy;
  verts[3 * v + 2] = tvz;
}

// ----------------- per-face coefficients + B fragments ----------------------
__global__ void face_setup_kernel(const float* __restrict__ vin,   // original vertices
                                  const int* __restrict__ faces,
                                  const float* __restrict__ verts, // screen verts
                                  float* __restrict__ fn,
                                  v2f* __restrict__ bfrag) {       // Bw0|Bw1|Bz
  int f = blockIdx.x * blockDim.x + threadIdx.x;
  if (f >= NB * 16) return;

  float w0x = 0.f, w0y = 0.f, w0c = -BIGF;   // default: face never wins
  float w1x = 0.f, w1y = 0.f, w1c = 0.f;
  float zx = 0.f, zy = 0.f, zc = 0.f;

  if (f < NFACE) {
    int i0 = faces[3 * f + 0], i1 = faces[3 * f + 1], i2 = faces[3 * f + 2];

    // face normal from ORIGINAL vertices: cross(v1-v0, v2-v0)
    float ax = vin[3 * i0], ay = vin[3 * i0 + 1], az = vin[3 * i0 + 2];
    float e1x = vin[3 * i1] - ax, e1y = vin[3 * i1 + 1] - ay, e1z = vin[3 * i1 + 2] - az;
    float e2x = vin[3 * i2] - ax, e2y = vin[3 * i2 + 1] - ay, e2z = vin[3 * i2 + 2] - az;
    fn[3 * f + 0] = e1y * e2z - e1z * e2y;
    fn[3 * f + 1] = e1z * e2x - e1x * e2z;
    fn[3 * f + 2] = e1x * e2y - e1y * e2x;

    // screen-space affine coefficients
    float x0 = verts[3 * i0], y0 = verts[3 * i0 + 1], z0 = verts[3 * i0 + 2];
    float x1 = verts[3 * i1], y1 = verts[3 * i1 + 1], z1 = verts[3 * i1 + 2];
    float x2 = verts[3 * i2], y2 = verts[3 * i2 + 1], z2 = verts[3 * i2 + 2];
    float area = (x1 - x0) * (y2 - y0) - (y1 - y0) * (x2 - x0);
    if (fabsf(area) > EPSA) {
      float inv = 1.0f / area;
      w0x = -(y2 - y1) * inv;
      w0y = (x2 - x1) * inv;
      w0c = ((y2 - y1) * x1 - (x2 - x1) * y1) * inv;
      w1x = -(y0 - y2) * inv;
      w1y = (x0 - x2) * inv;
      w1c = ((y0 - y2) * x2 - (x0 - x2) * y2) * inv;
      float dz0 = z0 - z2, dz1 = z1 - z2;
      zx = w0x * dz0 + w1x * dz1;
      zy = w0y * dz0 + w1y * dz1;
      zc = w0c * dz0 + w1c * dz1 + z2;
    }
  }

  // ISA 4x16 f32 B layout: lanes 0-15: V0=B[0][n], V1=B[1][n];
  //                        lanes 16-31: V0=B[2][n], V1=B[3][n]
  int bb = (f >> 4) * 32, n = f & 15;
  v2f lo, hi;
  lo.x = w0x; lo.y = w0y; hi.x = w0c; hi.y = 0.f;
  bfrag[bb + n] = lo; bfrag[bb + 16 + n] = hi;
  lo.x = w1x; lo.y = w1y; hi.x = w1c; hi.y = 0.f;
  bfrag[NB * 32 + bb + n] = lo; bfrag[NB * 32 + bb + 16 + n] = hi;
  lo.x = zx;  lo.y = zy;  hi.x = zc;  hi.y = 0.f;
  bfrag[2 * NB * 32 + bb + n] = lo; bfrag[2 * NB * 32 + bb + 16 + n] = hi;
}

// -------- deterministic vertex-normal gather (no float atomics) -------------
__global__ void vertex_normals_kernel(const int* __restrict__ faces,
                                      const float* __restrict__ fn,
                                      float* __restrict__ nrm) {
  int v = blockIdx.x * blockDim.x + threadIdx.x;
  if (v >= NVERT) return;
  float ax = 0.f, ay = 0.f, az = 0.f;
  for (int f = 0; f < NFACE; ++f) {
    int i0 = faces[3 * f + 0], i1 = faces[3 * f + 1], i2 = faces[3 * f + 2];
    int cnt = (i0 == v) + (i1 == v) + (i2 == v);
    if (cnt) {
      float w = (float)cnt;
      ax += w * fn[3 * f + 0];
      ay += w * fn[3 * f + 1];
      az += w * fn[3 * f + 2];
    }
  }
  float len = sqrtf(ax * ax + ay * ay + az * az);
  float r = 1.0f / fmaxf(len, 1e-6f);
  nrm[3 * v + 0] = ax * r;
  nrm[3 * v + 1] = ay * r;
  nrm[3 * v + 2] = az * r;
}

// ------------------------- WMMA rasterizer + shading ------------------------
__global__ __launch_bounds__(256) void raster_kernel(
    const float* __restrict__ bsrc,   // contiguous Bw0|Bw1|Bz (24000 floats)
    const int* __restrict__ faces, const float* __restrict__ nrm,
    float* __restrict__ img) {
  __shared__ float smem[BFLOATS];     // 96 KB of 320 KB WGP LDS

  const int tid = threadIdx.x;
  const int lane = tid & 31;

  // ---- stage the whole coefficient table once per block ----
#if USE_ASYNC_LDS
  for (int i = tid; i < BFLOATS / 4; i += 256) {
    __builtin_amdgcn_global_load_async_to_lds_b128(
        (gv4i_p)(bsrc + i * 4), (lv4i_p)(smem + i * 4),
        /*offset=*/0, /*cpol=*/0);
  }
  __builtin_amdgcn_s_wait_asynccnt(0);
  __syncthreads();
#else
  for (int i = tid; i < BFLOATS; i += 256) smem[i] = bsrc[i];
  __syncthreads();
#endif

  const int wave = (blockIdx.x * blockDim.x + tid) >> 5;
  const int row = wave >> 5;            // 32 strips of 16 pixels per row
  const int col0 = (wave & 31) << 4;
  if (row >= IMG) return;

  // A fragment (16x4 f32): lanes 0-15 -> (X_m, Y); lanes 16-31 -> (1, 0)
  float X = 1.0f - (2.0f * (float)(col0 + (lane & 15)) + 1.0f) * (1.0f / IMG);
  float Y = 1.0f - (2.0f * (float)row + 1.0f) * (1.0f / IMG);
  v2f a;
  if (lane < 16) { a.x = X; a.y = Y; } else { a.x = 1.0f; a.y = 0.0f; }

  const v2f* lb = (const v2f*)smem;

  float bZ[8], bU[8], bV[8];
  int bF[8];
#pragma unroll
  for (int r = 0; r < 8; ++r) { bZ[r] = BIGF; bF[r] = -1; bU[r] = 0.f; bV[r] = 0.f; }

  for (int b = 0; b < NB; ++b) {
    v2f f0 = lb[b * 32 + lane];
    v2f f1 = lb[NB * 32 + b * 32 + lane];
    v2f fz = lb[2 * NB * 32 + b * 32 + lane];
    v8f zero = {};
    v8f W0 = __builtin_amdgcn_wmma_f32_16x16x4_f32(false, a, false, f0, (short)0, zero, false, false);
    v8f W1 = __builtin_amdgcn_wmma_f32_16x16x4_f32(false, a, false, f1, (short)0, zero, false, false);
    v8f Zp = __builtin_amdgcn_wmma_f32_16x16x4_f32(false, a, false, fz, (short)0, zero, false, false);
    int face = b * 16 + (lane & 15);
#pragma unroll
    for (int r = 0; r < 8; ++r) {
      float u = W0[r], w = W1[r], z = Zp[r];
      // (u>=0 && w>=0 && 1-u-w>=0 && z>=0 && z<best) == (min3(u,w,z)>=0 && u+w<=1 && z<best)
      float m = fminf(fminf(u, w), z);
      bool ok = (m >= 0.f) & (u + w <= 1.f) & (z < bZ[r]);
      bZ[r] = ok ? z : bZ[r];
      bF[r] = ok ? face : bF[r];
      bU[r] = ok ? u : bU[r];
      bV[r] = ok ? w : bV[r];
    }
  }

  // Cross-lane min-z reduction within each 16-lane half; tie-break lowest face.
#pragma unroll
  for (int r = 0; r < 8; ++r) {
    float z = bZ[r], u = bU[r], w = bV[r];
    int f = bF[r];
#pragma unroll
    for (int m = 1; m <= 8; m <<= 1) {
      float zo = __shfl_xor(z, m, 32);
      int fo = __shfl_xor(f, m, 32);
      float uo = __shfl_xor(u, m, 32);
      float wo = __shfl_xor(w, m, 32);
      bool take = (zo < z) || ((zo == z) && (fo < f));
      z = take ? zo : z;
      f = take ? fo : f;
      u = take ? uo : u;
      w = take ? wo : w;
    }
    if ((lane & 15) == r) {
      int col = col0 + r + ((lane >> 4) << 3);  // high half owns pixel r+8
      int pix = row * IMG + col;
      float r0 = 1.0f, r1 = 1.0f, r2 = 1.0f;    // background = 1
      if (f >= 0) {
        int i0 = faces[3 * f + 0], i1 = faces[3 * f + 1], i2 = faces[3 * f + 2];
        float t = 1.0f - u - w;
        float nx = u * nrm[3 * i0 + 0] + w * nrm[3 * i1 + 0] + t * nrm[3 * i2 + 0];
        float ny = u * nrm[3 * i0 + 1] + w * nrm[3 * i1 + 1] + t * nrm[3 * i2 + 1];
        float nz = u * nrm[3 * i0 + 2] + w * nrm[3 * i1 + 2] + t * nrm[3 * i2 + 2];
        const float s3 = 0.57735026918962576f;
        float d0 = fminf(fmaxf(s3 * (-nx + ny + nz), 0.f), 1.f);
        float d1 = fminf(fmaxf(s3 * (nx + ny + nz), 0.f), 1.f);
        float d2 = fminf(fmaxf(s3 * (-nx - ny + nz), 0.f), 1.f);
        float d3 = fminf(fmaxf(s3 * (nx - ny + nz), 0.f), 1.f);
        float d4 = fminf(fmaxf(nz, 0.f), 1.f);
        float shading = (d0 + d1 + d2 + d3 + d4) * (1.7f / 5.0f);
        float sumw = u + w + t;
        r0 = (12.0f / 255.0f) * sumw * shading;
        r1 = (156.0f / 255.0f) * sumw * shading;
        r2 = (91.0f / 255.0f) * sumw * shading;
      }
      img[0 * IMG * IMG + pix] = r0;
      img[1 * IMG * IMG + pix] = r1;
      img[2 * IMG * IMG + pix] = r2;
    }
  }
}

// ----------------------------------------------------------------------------
extern "C" void kernel_launch(void* const* d_in, const int* in_sizes, int n_in,
                              void* d_out, int out_size, void* d_ws, size_t ws_size,
                              hipStream_t stream) {
  const float* vertices = (const float*)d_in[0];  // (1, 2048, 3)
  const float* cam = (const float*)d_in[1];       // (1, 3)
  const int* faces = (const int*)d_in[2];         // (2000, 3)

  float* out = (float*)d_out;
  float* img = out;                   // 3*512*512
  float* tv = out + 3 * IMG * IMG;    // 2048*3

  float* ws = (float*)d_ws;
  float* verts = ws;                  // 6144
  float* nrm = ws + 6144;             // 6144
  float* fn = ws + 12288;             // 6000
  float* bfrag = ws + 18288;          // 24000 floats (Bw0|Bw1|Bz), 16B-aligned

  project_kernel<<<(NVERT + 255) / 256, 256, 0, stream>>>(vertices, cam, tv, verts);
  face_setup_kernel<<<(NB * 16 + 255) / 256, 256, 0, stream>>>(vertices, faces, verts, fn,
                                                               (v2f*)bfrag);
  vertex_normals_kernel<<<(NVERT + 255) / 256, 256, 0, stream>>>(faces, fn, nrm);
  raster_kernel<<<(IMG * 32) / 8, 256, 0, stream>>>(bfrag, faces, nrm, img);
}